// SpatialConvBlock_80324478370218
// MI455X (gfx1250) — compile-verified
//
#include <hip/hip_runtime.h>

// CDNA5 / gfx1250 fused spatial-graph-conv block.
// Stage A: Y[192 x 200] = W[192x64] @ Xtile[64x200]  (+bias) via v_wmma_f32_16x16x4_f32
// Stage B: out[c,t,w] = sum_{k,v} Y[k*64+c, t*25+v] * A[k,v,w]  (VALU from LDS)
// Y never touches HBM; x-tile in and out-tile back are fully coalesced b128.

typedef __attribute__((ext_vector_type(2))) float v2f;
typedef __attribute__((ext_vector_type(8))) float v8f;

#define TT      8            // t rows per block
#define NV      25           // graph vertices
#define PVALID  (TT * NV)    // 200 valid spatial points per tile
#define NTILES  13           // ceil(200/16) WMMA N-tiles
#define XSTRIDE 208          // 13*16; xs row stride (disjoint banks for B-frag halves)
#define XCHUNKS (XSTRIDE / 4)
#define YSTRIDE 212          // ys row stride (conflict-free D-store + stage-B reads)
#define IN_C    64
#define OUT_CK  192          // 64 channels * 3 kernel partitions
#define T_LEN   300
#define NTHREADS 384

__global__ __launch_bounds__(NTHREADS)
void spatial_conv_block_fused(const float* __restrict__ x,
                              const float* __restrict__ A,
                              const float* __restrict__ W,
                              const float* __restrict__ bias,
                              float* __restrict__ out) {
  __shared__ __align__(16) float xs[IN_C * XSTRIDE];   //  53,248 B (reused as out-tile)
  __shared__ __align__(16) float ys[OUT_CK * YSTRIDE]; // 162,816 B
  __shared__ float as[3 * 625];                        //   7,500 B  (~223.6 KB total)

  const int tid = threadIdx.x;
  const int n   = blockIdx.y;
  const int t0  = blockIdx.x * TT;
  const int validp = min(PVALID, (T_LEN - t0) * NV);   // 200, or 100 on the tail block

  // ---- adjacency into LDS ----
  for (int i = tid; i < 3 * 625; i += NTHREADS) as[i] = A[i];

  // ---- x tile into LDS: each channel row is 200 contiguous floats in HBM ----
  for (int ch = tid; ch < IN_C * XCHUNKS; ch += NTHREADS) {
    const int ci = ch / XCHUNKS;
    const int cc = ch - ci * XCHUNKS;
    const int p0 = cc * 4;
    const float* gsrc = x + ((n * IN_C + ci) * T_LEN + t0) * NV;  // + p is contiguous
    float4 val = make_float4(0.f, 0.f, 0.f, 0.f);
    if (p0 + 3 < validp) {
      val = *(const float4*)(gsrc + p0);               // global_load_b128
    } else {
      if (p0 + 0 < validp) val.x = gsrc[p0 + 0];
      if (p0 + 1 < validp) val.y = gsrc[p0 + 1];
      if (p0 + 2 < validp) val.z = gsrc[p0 + 2];
      if (p0 + 3 < validp) val.w = gsrc[p0 + 3];
    }
    *(float4*)&xs[ci * XSTRIDE + p0] = val;            // ds_store_b128
  }

  // ---- per-wave W fragments (WMMA A-matrix, 16x4 f32 layout) ----
  // lanes 0-15: M=lane, K cols {0,1}; lanes 16-31: M=lane-16, K cols {2,3}
  const int lane  = tid & 31;
  const int wid   = tid >> 5;        // 0..11 -> one 16-channel M-tile
  const int lr    = lane & 15;
  const int kh    = lane >> 4;       // lane-half
  const int obase = wid * 16;

  v2f afrag[16];
#pragma unroll
  for (int kk = 0; kk < 16; ++kk) {
    const int col = kk * 4 + 2 * kh;
    const float* wp = W + (obase + lr) * IN_C + col;
    afrag[kk].x = wp[0];
    afrag[kk].y = wp[1];
  }
  float biasr[8];
#pragma unroll
  for (int r = 0; r < 8; ++r) biasr[r] = bias[obase + 8 * kh + r];

  __syncthreads();

  // ---- stage A: WMMA GEMM, every wave EXEC=all-ones ----
  for (int j = 0; j < NTILES; ++j) {
    const int pb = j * 16;
    v8f acc = {};
#pragma unroll
    for (int kk = 0; kk < 16; ++kk) {
      const int row = kk * 4 + 2 * kh;   // B-frag: V0 holds K rows {0,2}, V1 {1,3}
      v2f bfrag;
      bfrag.x = xs[row * XSTRIDE + pb + lr];
      bfrag.y = xs[(row + 1) * XSTRIDE + pb + lr];
      acc = __builtin_amdgcn_wmma_f32_16x16x4_f32(
          false, afrag[kk], false, bfrag, (short)0, acc, false, false);
    }
    // D layout: reg r, lane -> M = r + 8*kh, N = lr
#pragma unroll
    for (int r = 0; r < 8; ++r)
      ys[(obase + 8 * kh + r) * YSTRIDE + pb + lr] = acc[r] + biasr[r];
  }

  __syncthreads();   // ys ready; xs dead -> reuse xs as the staged out-tile

  // ---- stage B: out[c,tl,w] = sum_{k,v} ys[k*64+c][tl*25+v] * as[k,v,w] ----
  for (int pair = tid; pair < 64 * TT; pair += NTHREADS) {
    const int c  = pair >> 3;
    const int tl = pair & (TT - 1);
    float accw[NV];
#pragma unroll
    for (int w = 0; w < NV; ++w) accw[w] = 0.0f;
    for (int k = 0; k < 3; ++k) {
      const float* yrow = &ys[(k * 64 + c) * YSTRIDE + tl * NV];
      const float* arow = &as[k * 625];
      for (int v = 0; v < NV; ++v) {
        const float yv = yrow[v];
#pragma unroll
        for (int w = 0; w < NV; ++w)
          accw[w] = fmaf(yv, arow[v * NV + w], accw[w]);
      }
    }
    float* orow = &xs[c * XSTRIDE + tl * NV];          // stage into LDS
#pragma unroll
    for (int w = 0; w < NV; ++w) orow[w] = accw[w];
  }

  __syncthreads();

  // ---- coalesced out-tile store: each channel row is 200 contiguous floats ----
  for (int ch = tid; ch < 64 * (PVALID / 4); ch += NTHREADS) {
    const int c  = ch / (PVALID / 4);
    const int cc = ch - c * (PVALID / 4);
    const int p0 = cc * 4;
    float* gdst = out + ((n * 64 + c) * T_LEN + t0) * NV;  // + p is contiguous
    if (p0 + 3 < validp) {
      *(float4*)(gdst + p0) = *(const float4*)&xs[c * XSTRIDE + p0];  // b128
    } else {
      if (p0 + 0 < validp) gdst[p0 + 0] = xs[c * XSTRIDE + p0 + 0];
      if (p0 + 1 < validp) gdst[p0 + 1] = xs[c * XSTRIDE + p0 + 1];
      if (p0 + 2 < validp) gdst[p0 + 2] = xs[c * XSTRIDE + p0 + 2];
      if (p0 + 3 < validp) gdst[p0 + 3] = xs[c * XSTRIDE + p0 + 3];
    }
  }
}

extern "C" void kernel_launch(void* const* d_in, const int* in_sizes, int n_in,
                              void* d_out, int out_size, void* d_ws, size_t ws_size,
                              hipStream_t stream) {
  (void)in_sizes; (void)n_in; (void)out_size; (void)d_ws; (void)ws_size;
  const float* x = (const float*)d_in[0];   // [64,64,300,25]
  const float* A = (const float*)d_in[1];   // [3,25,25]
  const float* W = (const float*)d_in[2];   // [192,64,1,1]
  const float* b = (const float*)d_in[3];   // [192]
  float* out = (float*)d_out;               // out [64,64,300,25] ++ A [3,25,25]

  dim3 grid((T_LEN + TT - 1) / TT, 64);     // 38 x 64 blocks
  spatial_conv_block_fused<<<grid, NTHREADS, 0, stream>>>(x, A, W, b, out);

  // second tuple element: pass A through
  hipMemcpyAsync(out + 64 * 64 * T_LEN * NV, A, 3 * 625 * sizeof(float),
                 hipMemcpyDeviceToDevice, stream);
}